// PiolaKirchhoff_65231963291851
// MI455X (gfx1250) — compile-verified
//
#include <hip/hip_runtime.h>
#include <stdint.h>

// P = dW/dF for W = 8*I1 + 10*J^2 - 56*log(J) + 0.2*(I4^2 + I5^2) - 44
// with C = F^T (elementwise*) F  => C_ij = F_ij * F_ji (symmetric),
// I1 = sum F_ii^2, I4 = 4 F00^2 + .5 F11^2 + .5 F22^2, J = det(F),
// I5 = 4*(F11^2 F22^2 - F12^2 F21^2) + .5*(F00^2 F22^2 - F02^2 F20^2)
//    + .5*(F00^2 F11^2 - F01^2 F10^2).
// Analytic gradient assembled per entry below.

#define TPB 256
// Per block: 256 matrices * 9 floats * 4 B = 9216 B = 576 b128 transfers
//          = 2 full wavefront-wide passes (offset 0, 4096) + 1/4 pass (offset 8192, t<64).

__device__ __forceinline__ void pk_grad(const float f[9], float p[9]) {
    const float f00 = f[0], f01 = f[1], f02 = f[2];
    const float f10 = f[3], f11 = f[4], f12 = f[5];
    const float f20 = f[6], f21 = f[7], f22 = f[8];

    // cofactor(F) = dJ/dF
    const float a00 = f11 * f22 - f12 * f21;
    const float a01 = f12 * f20 - f10 * f22;
    const float a02 = f10 * f21 - f11 * f20;
    const float a10 = f02 * f21 - f01 * f22;
    const float a11 = f00 * f22 - f02 * f20;
    const float a12 = f01 * f20 - f00 * f21;
    const float a20 = f01 * f12 - f02 * f11;
    const float a21 = f02 * f10 - f00 * f12;
    const float a22 = f00 * f11 - f01 * f10;

    const float J = f00 * a00 + f01 * a01 + f02 * a02;

    const float s00 = f00 * f00, s01 = f01 * f01, s02 = f02 * f02;
    const float s10 = f10 * f10, s11 = f11 * f11, s12 = f12 * f12;
    const float s20 = f20 * f20, s21 = f21 * f21, s22 = f22 * f22;

    const float I4 = 4.0f * s00 + 0.5f * s11 + 0.5f * s22;
    const float I5 = 4.0f * (s11 * s22 - s12 * s21)
                   + 0.5f * (s00 * s22 - s02 * s20)
                   + 0.5f * (s00 * s11 - s01 * s10);

    const float cJ = 20.0f * J - 56.0f / J;  // dW/dJ
    const float c4 = 0.4f * I4;              // dW/dI4
    const float c5 = 0.4f * I5;              // dW/dI5

    p[0] = 16.0f * f00 + cJ * a00 + 8.0f * c4 * f00 + c5 * f00 * (s11 + s22);
    p[1] =               cJ * a01                   - c5 * f01 * s10;
    p[2] =               cJ * a02                   - c5 * f02 * s20;
    p[3] =               cJ * a10                   - c5 * f10 * s01;
    p[4] = 16.0f * f11 + cJ * a11 +        c4 * f11 + c5 * f11 * (8.0f * s22 + s00);
    p[5] =               cJ * a12                   - 8.0f * c5 * f12 * s21;
    p[6] =               cJ * a20                   - c5 * f20 * s02;
    p[7] =               cJ * a21                   - 8.0f * c5 * f21 * s12;
    p[8] = 16.0f * f22 + cJ * a22 +        c4 * f22 + c5 * f22 * (8.0f * s11 + s00);
}

__global__ __launch_bounds__(TPB) void PiolaKirchhoff_kernel(
    const float* __restrict__ F, float* __restrict__ P, int nMat) {
    __shared__ __align__(16) float tile[TPB * 9];  // 9216 B / block

    const int t          = threadIdx.x;
    const int blockFirst = (int)blockIdx.x * TPB;

    if (blockFirst + TPB <= nMat) {
        // ---- fast path: async global -> LDS staging (CDNA5 ASYNCcnt path) ----
        const uint32_t ldsAddr = ((uint32_t)(uintptr_t)&tile[0]) + (uint32_t)t * 16u;
        const uint32_t goff    = (uint32_t)t * 16u;
        {
            const uint64_t src = (uint64_t)(uintptr_t)(F + (size_t)blockFirst * 9);
            // IOFFSET is added to BOTH the LDS and the global address (ISA 08 §4.4),
            // so three static issues cover the whole 9216 B tile with no loop.
            asm volatile("global_load_async_to_lds_b128 %0, %1, %2"
                         :: "v"(ldsAddr), "v"(goff), "s"(src) : "memory");
            asm volatile("global_load_async_to_lds_b128 %0, %1, %2 offset:4096"
                         :: "v"(ldsAddr), "v"(goff), "s"(src) : "memory");
            if (t < 64)
                asm volatile("global_load_async_to_lds_b128 %0, %1, %2 offset:8192"
                             :: "v"(ldsAddr), "v"(goff), "s"(src) : "memory");
            asm volatile("s_wait_asynccnt 0" ::: "memory");
        }
        __syncthreads();

        // Thread t owns tile[t*9 .. t*9+8]: read F, compute P, write back in place.
        float f[9], p[9];
#pragma unroll
        for (int k = 0; k < 9; ++k) f[k] = tile[t * 9 + k];
        pk_grad(f, p);
#pragma unroll
        for (int k = 0; k < 9; ++k) tile[t * 9 + k] = p[k];

        __syncthreads();

        // ---- async LDS -> global drain, non-temporal so the output stream ----
        // ---- doesn't evict the (replay-resident) 144 MB input from L2      ----
        {
            const uint64_t dst = (uint64_t)(uintptr_t)(P + (size_t)blockFirst * 9);
            asm volatile("global_store_async_from_lds_b128 %0, %1, %2 th:TH_STORE_NT"
                         :: "v"(goff), "v"(ldsAddr), "s"(dst) : "memory");
            asm volatile("global_store_async_from_lds_b128 %0, %1, %2 offset:4096 th:TH_STORE_NT"
                         :: "v"(goff), "v"(ldsAddr), "s"(dst) : "memory");
            if (t < 64)
                asm volatile("global_store_async_from_lds_b128 %0, %1, %2 offset:8192 th:TH_STORE_NT"
                             :: "v"(goff), "v"(ldsAddr), "s"(dst) : "memory");
            asm volatile("s_wait_asynccnt 0" ::: "memory");
        }
    } else {
        // ---- guarded tail path (block-uniform branch; no barriers used) ----
        const int m = blockFirst + t;
        if (m < nMat) {
            float f[9], p[9];
#pragma unroll
            for (int k = 0; k < 9; ++k) f[k] = F[(size_t)m * 9 + k];
            pk_grad(f, p);
#pragma unroll
            for (int k = 0; k < 9; ++k) P[(size_t)m * 9 + k] = p[k];
        }
    }
}

extern "C" void kernel_launch(void* const* d_in, const int* in_sizes, int n_in,
                              void* d_out, int out_size, void* d_ws, size_t ws_size,
                              hipStream_t stream) {
    const float* F = (const float*)d_in[0];
    float* P       = (float*)d_out;
    const int nMat = in_sizes[0] / 9;  // F is [N,3,3] fp32
    const int grid = (nMat + TPB - 1) / TPB;
    PiolaKirchhoff_kernel<<<grid, TPB, 0, stream>>>(F, P, nMat);
}